// SelfConvergence_88957362635271
// MI455X (gfx1250) — compile-verified
//
#include <hip/hip_runtime.h>

typedef float v2f __attribute__((ext_vector_type(2)));
typedef float v8f __attribute__((ext_vector_type(8)));

#define LOG2E_F 1.4426950408889634f
#define INF_F   10000.0f

constexpr int NB = 8, NL = 512, ND = 64;

// One block per (b, d) column and j-pass. Block = 256 threads = 8 waves.
// Wave w handles j-tile [pass*128 + w*16, +16). Inner loop: i in chunks of 4,
// accumulated with V_WMMA_F32_16X16X4_F32:
//   A[m=jloc, k] = exp2( (v_i - v_j)^2 * (-w_j*log2e) )   (diag -> INF)
//   B[k, n=0]    = v_i   (numerator column)
//   B[k, n=1]    = 1.0   (denominator column)
// D[m,0] = sum_i e*v ; D[m,1] = sum_i e  (other columns ignored)
__global__ __launch_bounds__(256) void selfconv_attn_wmma(
    const float* __restrict__ x, const float* __restrict__ w,
    float* __restrict__ out) {
  __shared__ float sv[NL];   // x[b, :, d]
  __shared__ float swn[NL];  // -w[:, d] * log2(e)

  const int col = blockIdx.x;          // 0..511  == b*64 + d
  const int b   = col >> 6;
  const int d   = col & 63;
  const int tid = threadIdx.x;

  for (int l = tid; l < NL; l += 256) {
    sv[l]  = x[(b * NL + l) * ND + d];
    swn[l] = -w[l * ND + d] * LOG2E_F;
  }
  __syncthreads();

  const int lane = tid & 31;
  const int wave = tid >> 5;
  const int half = lane >> 4;          // 0: k=0,1   1: k=2,3
  const int jloc = lane & 15;          // A row / D column index
  const int ioff = half << 1;
  const bool denCol = (jloc == 1);     // this lane feeds the all-ones B column

  const int jbase = (int)blockIdx.y * 128 + wave * 16;
  const int jglob = jbase + jloc;
  const float vj = sv[jglob];
  const float wn = swn[jglob];

  v8f acc = {};
#pragma unroll 4
  for (int c = 0; c < NL; c += 4) {
    const int i0 = c + ioff;
    const float vi0 = sv[i0];
    const float vi1 = sv[i0 + 1];
    float d0 = vi0 - vj;
    float d1 = vi1 - vj;
    if (i0     == jglob) d0 = INF_F;   // self-mask, computed literally
    if (i0 + 1 == jglob) d1 = INF_F;
    v2f a, bm;
    a.x = __builtin_amdgcn_exp2f(d0 * d0 * wn);
    a.y = __builtin_amdgcn_exp2f(d1 * d1 * wn);
    bm.x = denCol ? 1.0f : vi0;
    bm.y = denCol ? 1.0f : vi1;
    acc = __builtin_amdgcn_wmma_f32_16x16x4_f32(false, a, false, bm,
                                                (short)0, acc, false, false);
  }

  // D layout: lane l, vgpr r -> element [M = r + 8*(l>>4), N = l&15].
  // N=0 column (numerator) in lanes 0/16; N=1 (denominator) in lanes 1/17.
#pragma unroll
  for (int r = 0; r < 8; ++r) {
    const float num = __shfl(acc[r], (lane & 16) | 0, 32);
    const float den = __shfl(acc[r], (lane & 16) | 1, 32);
    if (jloc == 0) {
      const int j = jbase + r + 8 * half;
      out[(b * NL + j) * ND + d] = num / den;
    }
  }
}

__global__ __launch_bounds__(256) void selfconv_copy_x(
    const float4* __restrict__ src, float4* __restrict__ dst, int n4) {
  const int i = blockIdx.x * blockDim.x + threadIdx.x;
  if (i < n4) dst[i] = src[i];
}

extern "C" void kernel_launch(void* const* d_in, const int* in_sizes, int n_in,
                              void* d_out, int out_size, void* d_ws, size_t ws_size,
                              hipStream_t stream) {
  const float* x = (const float*)d_in[0];      // [8, 512, 64]
  const float* w = (const float*)d_in[1];      // [512, 64]
  float* out = (float*)d_out;                  // [out | x] concatenated

  dim3 grid(NB * ND, 4);                       // 512 columns x 4 j-passes
  selfconv_attn_wmma<<<grid, 256, 0, stream>>>(x, w, out);

  const int n4 = NB * NL * ND / 4;             // 65536 float4
  selfconv_copy_x<<<(n4 + 255) / 256, 256, 0, stream>>>(
      (const float4*)x, (float4*)(out + NB * NL * ND), n4);
}